// HierarchicalPathNetworkLayer_59150289600979
// MI455X (gfx1250) — compile-verified
//
#include <hip/hip_runtime.h>
#include <math.h>

#define D 256        // feature dim (fixed by reference)
#define DV (D / 4)   // float4 chunks per row

typedef __attribute__((ext_vector_type(2))) float v2f;
typedef __attribute__((ext_vector_type(8))) float v8f;

// ---------------------------------------------------------------------------
// dst[i,:] = src[ca[i],:] * src[cb[i],:]   (float4 vectorized, one thread per
// float4 chunk; fully coalesced on the write, gathered 1KB-row reads)
// ---------------------------------------------------------------------------
__global__ __launch_bounds__(256) void k_gather_mul(
    const float4* __restrict__ src, const int* __restrict__ ca,
    const int* __restrict__ cb, float4* __restrict__ dst, int n) {
  long long idx = (long long)blockIdx.x * blockDim.x + threadIdx.x;
  if (idx >= (long long)n * DV) return;
  int row = (int)(idx >> 6);
  int c   = (int)(idx & (DV - 1));
  float4 a = src[(long long)ca[row] * DV + c];
  float4 b = src[(long long)cb[row] * DV + c];
  dst[idx] = make_float4(a.x * b.x, a.y * b.y, a.z * b.z, a.w * b.w);
}

// ---------------------------------------------------------------------------
// Fused top level: p = h[ca[i]] * h[cb[i]]  (the parent value, never stored)
// acc[ca[i]] += p; acc[cb[i]] += p          (hardware f32 atomics into L2)
// ---------------------------------------------------------------------------
__global__ __launch_bounds__(256) void k_prod_scatter(
    const float* __restrict__ h, float* __restrict__ acc,
    const int* __restrict__ ca, const int* __restrict__ cb, int n) {
  long long idx = (long long)blockIdx.x * blockDim.x + threadIdx.x;
  if (idx >= (long long)n * DV) return;
  int row = (int)(idx >> 6);
  int c4  = (int)(idx & (DV - 1)) * 4;
  long long ia = (long long)ca[row] * D + c4;
  long long ib = (long long)cb[row] * D + c4;
  float4 va = *(const float4*)(h + ia);
  float4 vb = *(const float4*)(h + ib);
  float p0 = va.x * vb.x, p1 = va.y * vb.y, p2 = va.z * vb.z, p3 = va.w * vb.w;
  unsafeAtomicAdd(acc + ia + 0, p0);
  unsafeAtomicAdd(acc + ia + 1, p1);
  unsafeAtomicAdd(acc + ia + 2, p2);
  unsafeAtomicAdd(acc + ia + 3, p3);
  unsafeAtomicAdd(acc + ib + 0, p0);
  unsafeAtomicAdd(acc + ib + 1, p1);
  unsafeAtomicAdd(acc + ib + 2, p2);
  unsafeAtomicAdd(acc + ib + 3, p3);
}

// ---------------------------------------------------------------------------
// acc[ca[i]] += src[i];  acc[cb[i]] += src[i]
// ---------------------------------------------------------------------------
__global__ __launch_bounds__(256) void k_scatter_add(
    const float* __restrict__ src, float* __restrict__ acc,
    const int* __restrict__ ca, const int* __restrict__ cb, int n) {
  long long idx = (long long)blockIdx.x * blockDim.x + threadIdx.x;
  if (idx >= (long long)n * DV) return;
  int row = (int)(idx >> 6);
  int c4  = (int)(idx & (DV - 1)) * 4;
  float4 v = *(const float4*)(src + (long long)row * D + c4);
  long long ia = (long long)ca[row] * D + c4;
  long long ib = (long long)cb[row] * D + c4;
  unsafeAtomicAdd(acc + ia + 0, v.x);
  unsafeAtomicAdd(acc + ia + 1, v.y);
  unsafeAtomicAdd(acc + ia + 2, v.z);
  unsafeAtomicAdd(acc + ia + 3, v.w);
  unsafeAtomicAdd(acc + ib + 0, v.x);
  unsafeAtomicAdd(acc + ib + 1, v.y);
  unsafeAtomicAdd(acc + ib + 2, v.z);
  unsafeAtomicAdd(acc + ib + 3, v.w);
}

// ---------------------------------------------------------------------------
// out = silu(H @ W + b) using V_WMMA_F32_16X16X4_F32.
// One wave32 per 16x16 output tile; K=256 -> 64 WMMA ops/tile.
// A layout (ISA 7.12.2, 32-bit A 16x4): lane<16 holds row M=lane, K={kb,kb+1};
// lane>=16 holds row M=lane-16, K={kb+2,kb+3}.  B mirrors with N=lane&15.
// C/D: VGPR r -> row m0+r+8*(lane>>4), col n0+(lane&15).
// Epilogue: sigmoid via v_rcp_f32 (1 ulp) instead of IEEE div chain, and
// non-temporal stores so the 51MB output doesn't evict H1/W from L2.
// ---------------------------------------------------------------------------
__global__ __launch_bounds__(256) void k_gemm_bias_silu(
    const float* __restrict__ H, const float* __restrict__ Wm,
    const float* __restrict__ bias, float* __restrict__ out, int M) {
  const int lane = threadIdx.x & 31;
  const int wave = threadIdx.x >> 5;
  const int tile = blockIdx.x * 8 + wave;
  const int nTiles = D / 16;                   // 16 column tiles
  const int mTile = tile / nTiles;
  const int nTile = tile - mTile * nTiles;
  const int m0 = mTile * 16;
  const int n0 = nTile * 16;
  if (m0 >= M) return;                          // uniform across the wave

  const int l15 = lane & 15;
  const int hs  = lane >> 4;                    // half-wave select (0/1)

  const float* __restrict__ arow = H + (long long)(m0 + l15) * D;
  const float* __restrict__ bcol = Wm + n0 + l15;

  v8f acc = {0.f, 0.f, 0.f, 0.f, 0.f, 0.f, 0.f, 0.f};
#pragma unroll 8
  for (int kb = 0; kb < D; kb += 4) {
    const int k = kb + 2 * hs;                  // even -> 8B-aligned v2f load
    v2f a = *(const v2f*)(arow + k);            // A[m,k], A[m,k+1]
    v2f b;
    b.x = bcol[(long long)k * D];               // W[k,   n]
    b.y = bcol[(long long)(k + 1) * D];         // W[k+1, n]
    // 8 args: (neg_a, A, neg_b, B, c_mod, C, reuse_a, reuse_b)
    acc = __builtin_amdgcn_wmma_f32_16x16x4_f32(
        false, a, false, b, (short)0, acc, false, false);
  }

  const float bv = bias[n0 + l15];
  float* __restrict__ ocol = out + n0 + l15;
#pragma unroll
  for (int r = 0; r < 8; ++r) {
    const int row = m0 + r + 8 * hs;
    float x = acc[r] + bv;
    // SiLU: x * sigmoid(x); v_exp_f32 + v_rcp_f32, no div_scale chain
    float s = x * __builtin_amdgcn_rcpf(1.0f + __expf(-x));
    __builtin_nontemporal_store(s, ocol + (long long)row * D);
  }
}

// ---------------------------------------------------------------------------
extern "C" void kernel_launch(void* const* d_in, const int* in_sizes, int n_in,
                              void* d_out, int out_size, void* d_ws, size_t ws_size,
                              hipStream_t stream) {
  const float* feat = (const float*)d_in[0];
  const int*   c2a  = (const int*)d_in[1];
  const int*   c2b  = (const int*)d_in[2];
  const int*   c3a  = (const int*)d_in[3];
  const int*   c3b  = (const int*)d_in[4];
  const int*   c4a  = (const int*)d_in[5];
  const int*   c4b  = (const int*)d_in[6];
  const float* W    = (const float*)d_in[7];
  const float* b    = (const float*)d_in[8];

  const int N1 = in_sizes[0] / D;
  const int N2 = in_sizes[1];
  const int N3 = in_sizes[3];
  const int N4 = in_sizes[5];

  // Workspace regions (floats). Peak = (N2 + 2*N3)*D*4 bytes (512 MB).
  float* ws    = (float*)d_ws;
  float* h2    = ws;                         // R0: N2*D
  float* h3    = h2 + (size_t)N2 * D;        // R1: N3*D
  float* h3new = h3 + (size_t)N3 * D;        // R2: N3*D
  float* h2new = h3;   // R1 reused once original h3 is dead
  float* h1    = h2;   // R0 reused once original h2 is dead

  auto nblk = [](long long elems) { return (int)((elems + 255) / 256); };

  // 1) h2 = feat[c2a] * feat[c2b]
  k_gather_mul<<<nblk((long long)N2 * DV), 256, 0, stream>>>(
      (const float4*)feat, c2a, c2b, (float4*)h2, N2);
  // 2) h3 = h2[c3a] * h2[c3b]
  k_gather_mul<<<nblk((long long)N3 * DV), 256, 0, stream>>>(
      (const float4*)h2, c3a, c3b, (float4*)h3, N3);
  // 3) h3new = h3  (separate read/accum buffers keep scatter race-free)
  hipMemcpyAsync(h3new, h3, (size_t)N3 * D * sizeof(float),
                 hipMemcpyDeviceToDevice, stream);
  // 4) fused: h4_i = h3[c4a]*h3[c4b]; h3new[c4a]+=h4_i; h3new[c4b]+=h4_i
  k_prod_scatter<<<nblk((long long)N4 * DV), 256, 0, stream>>>(
      h3, h3new, c4a, c4b, N4);
  // 5) h2new = h2   (original h3 in R1 dead after step 4)
  hipMemcpyAsync(h2new, h2, (size_t)N2 * D * sizeof(float),
                 hipMemcpyDeviceToDevice, stream);
  // 6) h2new[c3a] += h3new; h2new[c3b] += h3new
  k_scatter_add<<<nblk((long long)N3 * DV), 256, 0, stream>>>(
      h3new, h2new, c3a, c3b, N3);
  // 7) h1 = feat    (original h2 in R0 dead after step 5)
  hipMemcpyAsync(h1, feat, (size_t)N1 * D * sizeof(float),
                 hipMemcpyDeviceToDevice, stream);
  // 8) h1[c2a] += h2new; h1[c2b] += h2new
  k_scatter_add<<<nblk((long long)N2 * DV), 256, 0, stream>>>(
      h2new, h1, c2a, c2b, N2);
  // 9) out = silu(h1 @ W + b) — f32 WMMA, 8 waves (= 8 tiles) per block
  const int tiles = ((N1 + 15) / 16) * (D / 16);
  k_gemm_bias_silu<<<(tiles + 7) / 8, 256, 0, stream>>>(
      h1, W, b, (float*)d_out, N1);
}